// CapsLayer_20151986553393
// MI455X (gfx1250) — compile-verified
//
#include <hip/hip_runtime.h>
#include <math.h>

typedef float v2f __attribute__((ext_vector_type(2)));
typedef float v8f __attribute__((ext_vector_type(8)));

#define R_ROUTES 4096
#define N_CAPS   32
#define N_OUT    16
#define N_IN     16
#define BATCH    16
#define XB_STRIDE (R_ROUTES * N_IN)           // 65536 floats per batch row of x
#define W_R_STRIDE (N_CAPS * N_OUT * N_IN)    // 8192 floats per route of W
#define S_COLS   (N_CAPS * N_OUT)             // 512
#define S_ELEMS  (BATCH * S_COLS)             // 8192
#define RPW_S    16                            // routes per wave in s-pass

// ---- WMMA helpers: 16x16 f32 tile built from four V_WMMA_F32_16X16X4_F32 ----

// A matrix (16x4 f32) chunk kc covers K = 4kc..4kc+3.
// lanes 0-15: M=lane, K={4kc,4kc+1} in (x,y); lanes 16-31: K={4kc+2,4kc+3}.
__device__ __forceinline__ void load_A(const float* __restrict__ x, int r, int lane,
                                       v2f a[4]) {
    int m  = lane & 15;
    int ko = (lane >> 4) << 1;  // 0 or 2
    const float* p = x + (size_t)m * XB_STRIDE + r * N_IN + ko;
#pragma unroll
    for (int kc = 0; kc < 4; ++kc) {
        a[kc].x = p[4 * kc];
        a[kc].y = p[4 * kc + 1];
    }
}

// B matrix (4x16 f32) chunk for tile (capsule) c: B[k][n] = W[r, c, n, k]
// lanes 0-15: N=lane, K={4kc,4kc+1}; lanes 16-31: K={4kc+2,4kc+3}.
__device__ __forceinline__ void load_B(const float* __restrict__ W, int r, int c, int lane,
                                       v2f b[4]) {
    int n  = lane & 15;
    int ko = (lane >> 4) << 1;
    const float* p = W + (size_t)r * W_R_STRIDE + (c * N_OUT + n) * N_IN + ko;
#pragma unroll
    for (int kc = 0; kc < 4; ++kc) {
        b[kc].x = p[4 * kc];
        b[kc].y = p[4 * kc + 1];
    }
}

__device__ __forceinline__ v8f wmma_k16(const v2f a[4], const v2f b[4]) {
    v8f acc = {0.f, 0.f, 0.f, 0.f, 0.f, 0.f, 0.f, 0.f};
#pragma unroll
    for (int kc = 0; kc < 4; ++kc) {
        acc = __builtin_amdgcn_wmma_f32_16x16x4_f32(
            false, a[kc], false, b[kc], (short)0, acc, false, false);
    }
    return acc;
}

// ---- per-capsule softmax stats over routes: colmax[c], inv_denom[c] ----
__global__ void softmax_stats_kernel(const float* __restrict__ bij,
                                     float* __restrict__ colmax,
                                     float* __restrict__ inv_denom) {
    __shared__ float red[256];
    int c = blockIdx.x;
    int t = threadIdx.x;
    float mx = -INFINITY;
    for (int r = t; r < R_ROUTES; r += 256) mx = fmaxf(mx, bij[r * N_CAPS + c]);
    red[t] = mx;
    __syncthreads();
    for (int s2 = 128; s2 > 0; s2 >>= 1) {
        if (t < s2) red[t] = fmaxf(red[t], red[t + s2]);
        __syncthreads();
    }
    float cm = red[0];
    __syncthreads();
    float sum = 0.f;
    for (int r = t; r < R_ROUTES; r += 256) sum += __expf(bij[r * N_CAPS + c] - cm);
    red[t] = sum;
    __syncthreads();
    for (int s2 = 128; s2 > 0; s2 >>= 1) {
        if (t < s2) red[t] += red[t + s2];
        __syncthreads();
    }
    if (t == 0) {
        colmax[c]    = cm;
        inv_denom[c] = 1.0f / red[0];
    }
}

// ---- s-pass: s[b,c,o] += sum_r softmax(b_ij)[r,c] * u_hat[b,r,c,o] (WMMA recompute) ----
__global__ void s_pass_kernel(const float* __restrict__ x, const float* __restrict__ W,
                              const float* __restrict__ bij,
                              const float* __restrict__ colmax,
                              const float* __restrict__ inv_denom,
                              float* __restrict__ s) {
    int lane  = threadIdx.x & 31;
    int wave  = (blockIdx.x * blockDim.x + threadIdx.x) >> 5;
    int r0    = wave * RPW_S;
    int half8 = (lane >> 4) << 3;  // rows 0..7 or 8..15 of the D tile
    int o     = lane & 15;

    for (int c = 0; c < N_CAPS; ++c) {
        float cm   = colmax[c];
        float invd = inv_denom[c];
        v8f sacc = {0.f, 0.f, 0.f, 0.f, 0.f, 0.f, 0.f, 0.f};
        for (int rr = 0; rr < RPW_S; ++rr) {
            int r = r0 + rr;
            v2f a[4], b[4];
            load_A(x, r, lane, a);
            load_B(W, r, c, lane, b);
            v8f d = wmma_k16(a, b);
            float wgt = __expf(bij[r * N_CAPS + c] - cm) * invd;
#pragma unroll
            for (int vi = 0; vi < 8; ++vi) sacc[vi] += wgt * d[vi];
        }
#pragma unroll
        for (int vi = 0; vi < 8; ++vi) {
            int m = vi + half8;  // batch index
            atomicAdd(&s[m * S_COLS + c * N_OUT + o], sacc[vi]);
        }
    }
}

// ---- squash: v = s^2 * s / ((1 + s^2) * sqrt(s^2)), elementwise ----
__global__ void squash_kernel(const float* __restrict__ s, float* __restrict__ v) {
    int i = blockIdx.x * blockDim.x + threadIdx.x;
    if (i < S_ELEMS) {
        float sv = s[i];
        float sq = sv * sv;
        v[i] = sq * sv / ((1.0f + sq) * sqrtf(sq));
    }
}

// ---- a-pass: b_ij[r,c] += mean_b sum_o u_hat[b,r,c,o] * v[b,c,o] (WMMA recompute) ----
__global__ void a_pass_kernel(const float* __restrict__ x, const float* __restrict__ W,
                              const float* __restrict__ v, float* __restrict__ bij) {
    int lane  = threadIdx.x & 31;
    int r     = (blockIdx.x * blockDim.x + threadIdx.x) >> 5;  // one route per wave
    int half8 = (lane >> 4) << 3;
    int o     = lane & 15;

    v2f a[4];
    load_A(x, r, lane, a);

    for (int c = 0; c < N_CAPS; ++c) {
        v2f b[4];
        load_B(W, r, c, lane, b);
        v8f d = wmma_k16(a, b);
        float partial = 0.f;
#pragma unroll
        for (int vi = 0; vi < 8; ++vi) {
            int m = vi + half8;
            partial += d[vi] * v[m * S_COLS + c * N_OUT + o];
        }
        // reduce 32 lanes -> full sum over (b, o)
#pragma unroll
        for (int off = 16; off >= 1; off >>= 1) partial += __shfl_xor(partial, off, 32);
        if (lane == 0) bij[r * N_CAPS + c] += partial * (1.0f / BATCH);
    }
}

extern "C" void kernel_launch(void* const* d_in, const int* in_sizes, int n_in,
                              void* d_out, int out_size, void* d_ws, size_t ws_size,
                              hipStream_t stream) {
    const float* x = (const float*)d_in[0];  // (16, 4096, 16)
    const float* W = (const float*)d_in[1];  // (4096, 32, 16, 16)
    float* out = (float*)d_out;              // (16, 32, 16, 1) = 8192 f32

    char* ws = (char*)d_ws;
    float* bij    = (float*)ws;                                   // 4096*32 = 131072 f32
    float* colmax = (float*)(ws + (size_t)R_ROUTES * N_CAPS * 4); // 32
    float* invd   = colmax + N_CAPS;                              // 32
    float* sbuf   = invd + N_CAPS;                                // 8192
    float* vbuf   = sbuf + S_ELEMS;                               // 8192

    hipMemsetAsync(bij, 0, (size_t)R_ROUTES * N_CAPS * sizeof(float), stream);

    for (int it = 0; it < 3; ++it) {
        softmax_stats_kernel<<<N_CAPS, 256, 0, stream>>>(bij, colmax, invd);
        hipMemsetAsync(sbuf, 0, S_ELEMS * sizeof(float), stream);
        // 4096 routes / 16 per wave = 256 waves, 8 waves (256 thr) per block
        s_pass_kernel<<<(R_ROUTES / RPW_S) / 8, 256, 0, stream>>>(x, W, bij, colmax, invd, sbuf);
        float* vout = (it == 2) ? out : vbuf;
        squash_kernel<<<(S_ELEMS + 255) / 256, 256, 0, stream>>>(sbuf, vout);
        if (it < 2) {
            // 4096 routes, one wave each, 8 waves per block
            a_pass_kernel<<<R_ROUTES / 8, 256, 0, stream>>>(x, W, vbuf, bij);
        }
    }
}